// DimNet_90563680403816
// MI455X (gfx1250) — compile-verified
//
#include <hip/hip_runtime.h>
#include <hip/hip_bf16.h>

typedef __attribute__((ext_vector_type(2))) float v2f;
typedef __attribute__((ext_vector_type(8))) float v8f;

// ---------------- geometry constants ----------------
#define NPAD   540800      // 2*25*104*104
#define K1     2025
#define K1P    2028        // padded to multiple of 4
#define K2     4500        // already multiple of 4
#define NBUF   3600000     // 2*20*3*3*100*100
#define NW1P   (400*K1P)

// workspace byte offsets (256B aligned)
#define OFF_PADDED 0
#define OFF_W1P    2163200
#define OFF_T1     5408000
#define OFF_T2     5416192
#define OFF_BUF    5434368

// Build padded input: (2,25,104,104), border value 0.5
__global__ void pad_kernel(const float* __restrict__ pic, float* __restrict__ padded) {
    int i = blockIdx.x * 256 + threadIdx.x;
    if (i >= NPAD) return;
    int x = i % 104; int t = i / 104;
    int y = t % 104; t /= 104;               // t = b*25 + (a1*5+a2)
    float v = 0.5f;
    if (x >= 4 && x < 100 && y >= 4 && y < 100)
        v = pic[(t * 96 + (y - 4)) * 96 + (x - 4)];
    padded[i] = v;
}

// w1 (400 x 2025) -> zero-padded (400 x 2028) so the WMMA K-loop has no tail
__global__ void w1pad_kernel(const float* __restrict__ w1, float* __restrict__ w1p) {
    int i = blockIdx.x * 256 + threadIdx.x;
    if (i >= NW1P) return;
    int kk = i % K1P; int oc = i / K1P;
    w1p[i] = (kk < K1) ? w1[oc * K1 + kk] : 0.f;
}

// Gather-offset tables for the two implicit GEMMs
__global__ void tables_kernel(int* __restrict__ off1, int* __restrict__ off2) {
    int i = blockIdx.x * 256 + threadIdx.x;
    if (i < K1P) {
        int v = 0;                           // padded k -> offset 0 (weight is 0)
        if (i < K1) {
            int kw = i % 9; int t = i / 9;
            int kh = t % 9; t /= 9;
            int a2 = t % 5; int a1 = t / 5;
            v = ((a1 * 5 + a2) * 104 + kh) * 104 + kw;
        }
        off1[i] = v;
    }
    if (i < K2) {
        int kw2 = i % 5; int t = i / 5;
        int kh2 = t % 5; t /= 5;
        int kw1 = t % 3; t /= 3;
        int kh1 = t % 3; int mid = t / 3;
        off2[i] = ((mid * 9 + kh1 * 3 + kw1) * 100 + kh2) * 100 + kw2;
    }
}

// conv2a: buf(b,mid,i,j,y,x) = relu( sum_{p,q,u,v} padded[b,i+p,j+q,y+u,x+v]*w2a + b2a )
// 1.6% of total FLOPs -> plain VALU kernel is fine.
__global__ void conv2a_kernel(const float* __restrict__ padded, const float* __restrict__ w2a,
                              const float* __restrict__ b2a, float* __restrict__ buf) {
    int i = blockIdx.x * 256 + threadIdx.x;
    if (i >= NBUF) return;
    int x = i % 100; int t = i / 100;
    int y = t % 100; t /= 100;
    int j  = t % 3;  t /= 3;
    int ii = t % 3;  t /= 3;
    int mid = t % 20; int b = t / 20;
    float acc = b2a[mid];
    const float* wp = w2a + mid * 225;
    for (int p = 0; p < 3; ++p)
        for (int q = 0; q < 3; ++q) {
            const float* xp = padded + ((b * 25 + (ii + p) * 5 + (j + q)) * 104 + y) * 104 + x;
            const float* wq = wp + (p * 3 + q) * 25;
            #pragma unroll
            for (int u = 0; u < 5; ++u)
                #pragma unroll
                for (int v = 0; v < 5; ++v)
                    acc += xp[u * 104 + v] * wq[u * 5 + v];
        }
    buf[i] = acc > 0.f ? acc : 0.f;
}

// Fused implicit-GEMM, M-block=5, N-block=2: one wave computes 80(oc) x 32(pos).
// One wave per workgroup (blockDim=32) so all tile coordinates are derived from
// blockIdx only -> compiler proves them uniform -> offset-table reads become
// s_load_b128 and operand loads use the SADDR+voffset global form.
// WMMA f32 16x16x4 per-lane layout (hl = lane>>4, idx = lane&15):
//   A: a[0]=A[idx][k+2*hl], a[1]=A[idx][k+2*hl+1]
//   B: b[0]=B[k+2*hl][idx], b[1]=B[k+2*hl+1][idx]
//   C/D: element v of v8f = row (v + 8*hl), col idx
__global__ __launch_bounds__(32) void fused_wmma_gemm(
    const float* __restrict__ padded, const float* __restrict__ w1p,
    const int* __restrict__ off1, const float* __restrict__ b1v,
    const float* __restrict__ buf, const float* __restrict__ w2b,
    const int* __restrict__ off2, const float* __restrict__ b2bv,
    float* __restrict__ out) {
    const int lane = threadIdx.x;         // 0..31
    const int hl  = lane >> 4;
    const int idx = lane & 15;

    const int w = blockIdx.x;             // 0..2879 (uniform)
    const int posGrp = w % 288;
    const int r0 = w / 288;
    const int ocGrp = r0 % 5;
    const int b = r0 / 5;

    const int pos0 = posGrp * 32;         // 32 positions, never crosses a row (96%32==0)
    const int oh = pos0 / 96;
    const int ow0 = pos0 - oh * 96;
    const int ocBase = ocGrp * 80;

    // per-lane 32-bit element offsets (uniform 64-bit bases stay in SGPRs)
    const int pb0 = oh * 100 + ow0 + idx;           // into buf image
    const int pp0 = oh * 104 + ow0 + idx;           // into padded image
    const float* __restrict__ buf2 = buf + b * 1800000;
    const float* __restrict__ pad2 = padded + b * 270400;

    const int aoff2 = (ocBase + idx) * K2;          // w2b row for A operand
    const int aoff1 = (ocBase + idx) * K1P;         // w1p row for A operand
    const int hl2 = hl * 2;

    v8f acc2[5][2] = {};
    for (int k = 0; k < K2; k += 4) {
        const int4 q = *(const int4*)(off2 + k);    // uniform -> s_load_b128
        const int t0 = hl ? q.z : q.x;
        const int t1 = hl ? q.w : q.y;
        v2f bA, bB;
        bA.x = buf2[pb0 + t0];      bA.y = buf2[pb0 + t1];
        bB.x = buf2[pb0 + 16 + t0]; bB.y = buf2[pb0 + 16 + t1];
        const int kk = k + hl2;
        #pragma unroll
        for (int j = 0; j < 5; ++j) {
            v2f a = *(const v2f*)(w2b + aoff2 + j * (16 * K2) + kk);
            acc2[j][0] = __builtin_amdgcn_wmma_f32_16x16x4_f32(false, a, false, bA,
                                                    (short)0, acc2[j][0], false, false);
            acc2[j][1] = __builtin_amdgcn_wmma_f32_16x16x4_f32(false, a, false, bB,
                                                    (short)0, acc2[j][1], false, false);
        }
    }

    v8f acc1[5][2] = {};
    for (int k = 0; k < K1P; k += 4) {
        const int4 q = *(const int4*)(off1 + k);    // uniform -> s_load_b128
        const int t0 = hl ? q.z : q.x;
        const int t1 = hl ? q.w : q.y;
        v2f bA, bB;
        bA.x = pad2[pp0 + t0];      bA.y = pad2[pp0 + t1];
        bB.x = pad2[pp0 + 16 + t0]; bB.y = pad2[pp0 + 16 + t1];
        const int kk = k + hl2;
        #pragma unroll
        for (int j = 0; j < 5; ++j) {
            v2f a = *(const v2f*)(w1p + aoff1 + j * (16 * K1P) + kk);
            acc1[j][0] = __builtin_amdgcn_wmma_f32_16x16x4_f32(false, a, false, bA,
                                                    (short)0, acc1[j][0], false, false);
            acc1[j][1] = __builtin_amdgcn_wmma_f32_16x16x4_f32(false, a, false, bB,
                                                    (short)0, acc1[j][1], false, false);
        }
    }

    // ---- epilogue: bias, relu(p2), average, pixel-shuffle store ----
    float* __restrict__ outb = out + b * (25 * 384 * 384);
    #pragma unroll
    for (int j = 0; j < 5; ++j) {
        #pragma unroll
        for (int v = 0; v < 8; ++v) {
            const int oc = ocBase + j * 16 + v + hl * 8;
            const float bias2 = b2bv[oc];
            const float bias1 = b1v[oc];
            const int c25 = oc >> 4;
            const int rr = (oc >> 2) & 3;
            const int ss = oc & 3;
            #pragma unroll
            for (int n = 0; n < 2; ++n) {
                float p2 = acc2[j][n][v] + bias2;
                p2 = p2 > 0.f ? p2 : 0.f;
                const float val = (acc1[j][n][v] + bias1 + p2) * 0.5f;
                const int ow = ow0 + n * 16 + idx;
                outb[((c25 * 384) + oh * 4 + rr) * 384 + ow * 4 + ss] = val;
            }
        }
    }
}

extern "C" void kernel_launch(void* const* d_in, const int* in_sizes, int n_in,
                              void* d_out, int out_size, void* d_ws, size_t ws_size,
                              hipStream_t stream) {
    const float* pic = (const float*)d_in[0];
    const float* w1  = (const float*)d_in[1];
    const float* b1  = (const float*)d_in[2];
    const float* w2a = (const float*)d_in[3];
    const float* b2a = (const float*)d_in[4];
    const float* w2b = (const float*)d_in[5];
    const float* b2b = (const float*)d_in[6];
    float* out = (float*)d_out;
    char* ws = (char*)d_ws;

    float* padded = (float*)(ws + OFF_PADDED);
    float* w1p    = (float*)(ws + OFF_W1P);
    int*   off1   = (int*)(ws + OFF_T1);
    int*   off2   = (int*)(ws + OFF_T2);
    float* buf    = (float*)(ws + OFF_BUF);

    pad_kernel   <<<(NPAD  + 255) / 256, 256, 0, stream>>>(pic, padded);
    w1pad_kernel <<<(NW1P  + 255) / 256, 256, 0, stream>>>(w1, w1p);
    tables_kernel<<<(K2    + 255) / 256, 256, 0, stream>>>(off1, off2);
    conv2a_kernel<<<(NBUF  + 255) / 256, 256, 0, stream>>>(padded, w2a, b2a, buf);

    // 2880 waves: 2 batch x 5 oc-groups(80) x 288 pos-groups(32); 1 wave/block
    fused_wmma_gemm<<<2880, 32, 0, stream>>>(padded, w1p, off1, b1,
                                             buf, w2b, off2, b2b, out);
}